// GAT_20289425506879
// MI455X (gfx1250) — compile-verified
//
#include <hip/hip_runtime.h>
#include <math.h>

typedef __attribute__((ext_vector_type(2))) float v2f;
typedef __attribute__((ext_vector_type(4))) float v4f;
typedef __attribute__((ext_vector_type(8))) float v8f;

#define F_IN 512
#define HID 16
#define NEG_SLOPE 0.2f
#define EPS_DEN 1e-16f

// Monotonic float<->uint mapping so we can use integer atomicMax for segment_max.
__device__ __forceinline__ unsigned ord_of_float(float f) {
  unsigned u = __float_as_uint(f);
  return (u & 0x80000000u) ? ~u : (u | 0x80000000u);
}
__device__ __forceinline__ float float_of_ord(unsigned u) {
  return (u & 0x80000000u) ? __uint_as_float(u & 0x7FFFFFFFu) : __uint_as_float(~u);
}

// -------- edge_index int64 -> int32, append self-loops ----------------------
__global__ void gat_convert(const long long* __restrict__ ei, int* __restrict__ src,
                            int* __restrict__ dst, int e, int n) {
  int i = blockIdx.x * blockDim.x + threadIdx.x;
  int et = e + n;
  if (i >= et) return;
  if (i < e) {
    src[i] = (int)ei[i];
    dst[i] = (int)ei[(size_t)e + i];
  } else {
    src[i] = i - e;
    dst[i] = i - e;
  }
}

// -------- Layer-1 GEMM: h1 = x @ W1  (M=N, K=512, Nc=16) --------------------
// One wave per 16-row tile. x-tile moved global->LDS with the CDNA5 async
// copy path (GLOBAL_LOAD_ASYNC_TO_LDS_B128, ASYNCcnt), padded LDS stride 68
// -> conflict-free ds_load_b64 A reads. 128 x v_wmma_f32_16x16x4_f32 per tile.
__global__ __launch_bounds__(32) void gat_gemm1(const float* __restrict__ x,
                                                const float* __restrict__ W1,
                                                float* __restrict__ h1, int n) {
  __shared__ float ldsA[16 * 68];
  const int lane = threadIdx.x;
  const int l16 = lane & 15;
  const int half = lane >> 4;
  const int r0 = blockIdx.x * 16;
  v8f c = {};
  for (int k0 = 0; k0 < F_IN; k0 += 64) {
    // stage 16x64 tile of x: 8 x B128 async global->LDS copies per lane set
#pragma unroll
    for (int i = 0; i < 8; ++i) {
      int f = i * 32 + lane;
      int row = f >> 4;
      int cv = f & 15;
      int gr = r0 + row;
      if (gr >= n) gr = n - 1;
      const float* gp = x + (size_t)gr * F_IN + k0 + cv * 4;
      // low 32 bits of a generic LDS pointer == LDS byte offset (ISA aperture map)
      unsigned lp = (unsigned)(size_t)(&ldsA[row * 68 + cv * 4]);
      asm volatile("global_load_async_to_lds_b128 %0, %1, off"
                   :
                   : "v"(lp), "v"(gp)
                   : "memory");
    }
    asm volatile("s_wait_asynccnt 0x0" ::: "memory");
    __syncthreads();
#pragma unroll
    for (int j = 0; j < 16; ++j) {
      const int kk = 4 * j;
      v2f a, b;
      // A 16x4 layout: lane l16 = row M; half selects K pair {0,1} vs {2,3}
      a.x = ldsA[l16 * 68 + kk + 2 * half];
      a.y = ldsA[l16 * 68 + kk + 2 * half + 1];
      // B 4x16 layout: v0 = B[2*half][l16], v1 = B[2*half+1][l16]
      const float* bp = W1 + (size_t)(k0 + kk + 2 * half) * HID + l16;
      b.x = bp[0];
      b.y = bp[HID];
      c = __builtin_amdgcn_wmma_f32_16x16x4_f32(false, a, false, b, (short)0, c,
                                                false, false);
    }
    __syncthreads();
  }
  // C layout: reg j -> row r0 + j + 8*half, col l16
#pragma unroll
  for (int j = 0; j < 8; ++j) {
    int row = r0 + j + 8 * half;
    if (row < n) h1[(size_t)row * HID + l16] = c[j];
  }
}

// -------- Layer-2 GEMM: h2 = relu(agg1 + b1) @ W2  (K=16, Nc=16) ------------
__global__ __launch_bounds__(32) void gat_gemm2(const float* __restrict__ agg1,
                                                const float* __restrict__ b1,
                                                const float* __restrict__ W2,
                                                float* __restrict__ h2, int n) {
  const int lane = threadIdx.x;
  const int l16 = lane & 15;
  const int half = lane >> 4;
  const int r0 = blockIdx.x * 16;
  int ga = r0 + l16;
  if (ga >= n) ga = n - 1;
  v8f c = {};
#pragma unroll
  for (int j = 0; j < 4; ++j) {
    const int ka = 4 * j + 2 * half;
    v2f a, b;
    a.x = fmaxf(agg1[(size_t)ga * HID + ka] + b1[ka], 0.0f);
    a.y = fmaxf(agg1[(size_t)ga * HID + ka + 1] + b1[ka + 1], 0.0f);
    b.x = W2[ka * HID + l16];
    b.y = W2[(ka + 1) * HID + l16];
    c = __builtin_amdgcn_wmma_f32_16x16x4_f32(false, a, false, b, (short)0, c,
                                              false, false);
  }
#pragma unroll
  for (int j = 0; j < 8; ++j) {
    int row = r0 + j + 8 * half;
    if (row < n) h2[(size_t)row * HID + l16] = c[j];
  }
}

// -------- per-node attention logits: as = h.a_src, ad = h.a_dst -------------
__global__ void gat_logits(const float* __restrict__ h, const float* __restrict__ avs,
                           const float* __restrict__ avd, float* __restrict__ as_,
                           float* __restrict__ ad_, int n) {
  int i = blockIdx.x * blockDim.x + threadIdx.x;
  if (i >= n) return;
  const float* row = h + (size_t)i * HID;
  float s = 0.f, d = 0.f;
#pragma unroll
  for (int c2 = 0; c2 < HID; ++c2) {
    float v = row[c2];
    s += v * avs[c2];
    d += v * avd[c2];
  }
  as_[i] = s;
  ad_[i] = d;
}

// -------- init segment-max/sum/agg buffers ----------------------------------
__global__ void gat_init(unsigned* __restrict__ maxu, float* __restrict__ sum,
                         float* __restrict__ agg, int n) {
  int i = blockIdx.x * blockDim.x + threadIdx.x;
  if (i < n) {
    maxu[i] = 0u;  // ordered-uint floor (below any finite float's code)
    sum[i] = 0.f;
  }
  if (i < n * HID) agg[i] = 0.f;
}

__device__ __forceinline__ float edge_logit(const float* as_, const float* ad_,
                                            int s, int d) {
  float v = as_[s] + ad_[d];
  return v > 0.f ? v : NEG_SLOPE * v;
}

// -------- softmax pass 1: segment max over dst ------------------------------
__global__ void gat_edge_max(const int* __restrict__ src, const int* __restrict__ dst,
                             const float* __restrict__ as_, const float* __restrict__ ad_,
                             unsigned* __restrict__ maxu, int e) {
  int i = blockIdx.x * blockDim.x + threadIdx.x;
  if (i >= e) return;
  int d = dst[i];
  float v = edge_logit(as_, ad_, src[i], d);
  atomicMax(&maxu[d], ord_of_float(v));
}

// -------- softmax pass 2: segment sum of exp(e - max) -----------------------
__global__ void gat_edge_sum(const int* __restrict__ src, const int* __restrict__ dst,
                             const float* __restrict__ as_, const float* __restrict__ ad_,
                             const unsigned* __restrict__ maxu, float* __restrict__ sum,
                             int e) {
  int i = blockIdx.x * blockDim.x + threadIdx.x;
  if (i >= e) return;
  int d = dst[i];
  float v = edge_logit(as_, ad_, src[i], d);
  atomicAdd(&sum[d], __expf(v - float_of_ord(maxu[d])));
}

// -------- softmax pass 3: weighted aggregation, 16 lanes per edge -----------
__global__ void gat_edge_agg(const int* __restrict__ src, const int* __restrict__ dst,
                             const float* __restrict__ as_, const float* __restrict__ ad_,
                             const unsigned* __restrict__ maxu, const float* __restrict__ sum,
                             const float* __restrict__ h, float* __restrict__ agg, int e) {
  int t = blockIdx.x * blockDim.x + threadIdx.x;
  int i = t >> 4;
  int ch = t & 15;
  if (i >= e) return;
  int s = src[i], d = dst[i];
  float v = edge_logit(as_, ad_, s, d);
  float alpha = __expf(v - float_of_ord(maxu[d])) / (sum[d] + EPS_DEN);
  atomicAdd(&agg[(size_t)d * HID + ch], h[(size_t)s * HID + ch] * alpha);
}

// -------- out = log_softmax(agg2 + b2) --------------------------------------
__global__ void gat_final(const float* __restrict__ agg2, const float* __restrict__ b2,
                          float* __restrict__ out, int n) {
  int i = blockIdx.x * blockDim.x + threadIdx.x;
  if (i >= n) return;
  float r[HID];
  float m = -INFINITY;
#pragma unroll
  for (int c2 = 0; c2 < HID; ++c2) {
    r[c2] = agg2[(size_t)i * HID + c2] + b2[c2];
    m = fmaxf(m, r[c2]);
  }
  float s = 0.f;
#pragma unroll
  for (int c2 = 0; c2 < HID; ++c2) s += expf(r[c2] - m);
  float lse = m + logf(s);
#pragma unroll
  for (int c2 = 0; c2 < HID; ++c2) out[(size_t)i * HID + c2] = r[c2] - lse;
}

extern "C" void kernel_launch(void* const* d_in, const int* in_sizes, int n_in,
                              void* d_out, int out_size, void* d_ws, size_t ws_size,
                              hipStream_t stream) {
  (void)n_in; (void)out_size; (void)ws_size;
  const float* x      = (const float*)d_in[0];
  const float* W1     = (const float*)d_in[1];
  const float* a_src1 = (const float*)d_in[2];
  const float* a_dst1 = (const float*)d_in[3];
  const float* b1     = (const float*)d_in[4];
  const float* W2     = (const float*)d_in[5];
  const float* a_src2 = (const float*)d_in[6];
  const float* a_dst2 = (const float*)d_in[7];
  const float* b2     = (const float*)d_in[8];
  const long long* ei = (const long long*)d_in[9];
  float* out = (float*)d_out;

  const int n  = in_sizes[0] / F_IN;  // 100000
  const int e  = in_sizes[9] / 2;     // 3200000
  const int et = e + n;               // + self loops

  // workspace carve (reuse hbuf for h1 then h2, aggbuf for agg1 then agg2)
  char* w = (char*)d_ws;
  auto carve = [&](size_t bytes) {
    char* p = w;
    w += (bytes + 255) & ~(size_t)255;
    return p;
  };
  int* srcb      = (int*)carve((size_t)et * 4);
  int* dstb      = (int*)carve((size_t)et * 4);
  float* hbuf    = (float*)carve((size_t)n * HID * 4);
  float* aggbuf  = (float*)carve((size_t)n * HID * 4);
  float* asb     = (float*)carve((size_t)n * 4);
  float* adb     = (float*)carve((size_t)n * 4);
  unsigned* maxu = (unsigned*)carve((size_t)n * 4);
  float* sumb    = (float*)carve((size_t)n * 4);

  const int T = 256;
  const int tiles = (n + 15) / 16;

  gat_convert<<<(et + T - 1) / T, T, 0, stream>>>(ei, srcb, dstb, e, n);

  // ---- layer 1 ----
  gat_gemm1<<<tiles, 32, 0, stream>>>(x, W1, hbuf, n);
  gat_logits<<<(n + T - 1) / T, T, 0, stream>>>(hbuf, a_src1, a_dst1, asb, adb, n);
  gat_init<<<(n * HID + T - 1) / T, T, 0, stream>>>(maxu, sumb, aggbuf, n);
  gat_edge_max<<<(et + T - 1) / T, T, 0, stream>>>(srcb, dstb, asb, adb, maxu, et);
  gat_edge_sum<<<(et + T - 1) / T, T, 0, stream>>>(srcb, dstb, asb, adb, maxu, sumb, et);
  gat_edge_agg<<<(et * 16 + T - 1) / T, T, 0, stream>>>(srcb, dstb, asb, adb, maxu,
                                                        sumb, hbuf, aggbuf, et);

  // ---- layer 2 (relu+bias fused into GEMM A load; hbuf now holds h2) ----
  gat_gemm2<<<tiles, 32, 0, stream>>>(aggbuf, b1, W2, hbuf, n);
  gat_logits<<<(n + T - 1) / T, T, 0, stream>>>(hbuf, a_src2, a_dst2, asb, adb, n);
  gat_init<<<(n * HID + T - 1) / T, T, 0, stream>>>(maxu, sumb, aggbuf, n);
  gat_edge_max<<<(et + T - 1) / T, T, 0, stream>>>(srcb, dstb, asb, adb, maxu, et);
  gat_edge_sum<<<(et + T - 1) / T, T, 0, stream>>>(srcb, dstb, asb, adb, maxu, sumb, et);
  gat_edge_agg<<<(et * 16 + T - 1) / T, T, 0, stream>>>(srcb, dstb, asb, adb, maxu,
                                                        sumb, hbuf, aggbuf, et);

  gat_final<<<(n + T - 1) / T, T, 0, stream>>>(aggbuf, b2, out, n);
}